// Trainer_12335146074567
// MI455X (gfx1250) — compile-verified
//
#include <hip/hip_runtime.h>
#include <hip/hip_bf16.h>
#include <stdint.h>

#define N_RAYS      131072
#define N_SAMPLES   128
#define N_IMP       64
#define OUT_STRIDE  69
#define S_INV       64.0f

typedef float v2f __attribute__((ext_vector_type(2)));
typedef float v8f __attribute__((ext_vector_type(8)));

// ---------------------------------------------------------------------------
// Kernel 1: one wave32 per ray.
//   sigmoid -> alpha -> cumprod (shfl scan) -> weights -> rgb/depth/acc
//   writes normalized pdf[R,128] (slot 127 padded 0) to workspace (lives in L2)
// ---------------------------------------------------------------------------
__global__ __launch_bounds__(128) void k_weights(
    const float* __restrict__ sdf, const float* __restrict__ zv,
    const float* __restrict__ rad, float* __restrict__ out,
    float* __restrict__ pdfw)
{
    const int wave = threadIdx.x >> 5;
    const int lane = threadIdx.x & 31;
    const int ray  = blockIdx.x * 4 + wave;
    const int k0   = lane * 4;                       // blocked: lane holds k0..k0+3

    const float4 s4 = *reinterpret_cast<const float4*>(sdf + (size_t)ray * N_SAMPLES + k0);
    float c0 = 1.0f / (1.0f + __expf(-S_INV * s4.x));
    float c1 = 1.0f / (1.0f + __expf(-S_INV * s4.y));
    float c2 = 1.0f / (1.0f + __expf(-S_INV * s4.z));
    float c3 = 1.0f / (1.0f + __expf(-S_INV * s4.w));
    float cN = __shfl_down(c0, 1, 32);               // neighbor lane's first cdf

    // alpha_k = relu((cdf_k - cdf_{k+1}) / (cdf_k + 1e-10)), k = 0..126
    float a0 = fmaxf(0.0f, (c0 - c1) / (c0 + 1e-10f));
    float a1 = fmaxf(0.0f, (c1 - c2) / (c1 + 1e-10f));
    float a2 = fmaxf(0.0f, (c2 - c3) / (c2 + 1e-10f));
    float a3 = (lane == 31) ? 0.0f : fmaxf(0.0f, (c3 - cN) / (c3 + 1e-10f));

    float sh0 = 1.0f - a0 + 1e-10f;
    float sh1 = 1.0f - a1 + 1e-10f;
    float sh2 = 1.0f - a2 + 1e-10f;
    float sh3 = (lane == 31) ? 1.0f : (1.0f - a3 + 1e-10f);

    // multiplicative exclusive scan across the wave (Hillis-Steele)
    float incl = sh0 * sh1 * sh2 * sh3;
    #pragma unroll
    for (int off = 1; off < 32; off <<= 1) {
        float v = __shfl_up(incl, off, 32);
        if (lane >= off) incl *= v;
    }
    float excl = __shfl_up(incl, 1, 32);
    if (lane == 0) excl = 1.0f;

    float T0 = excl, T1 = T0 * sh0, T2 = T1 * sh1, T3 = T2 * sh2;
    float w0 = a0 * T0, w1 = a1 * T1, w2 = a2 * T2, w3 = a3 * T3;

    const float4 z4 = *reinterpret_cast<const float4*>(zv + (size_t)ray * N_SAMPLES + k0);
    float zN = __shfl_down(z4.x, 1, 32);
    float m0 = 0.5f * (z4.x + z4.y);
    float m1 = 0.5f * (z4.y + z4.z);
    float m2 = 0.5f * (z4.z + z4.w);
    float m3 = 0.5f * (z4.w + zN);

    const float* rp = rad + ((size_t)ray * (N_SAMPLES - 1) + k0) * 3;
    float pr = w0 * rp[0] + w1 * rp[3] + w2 * rp[6];
    float pg = w0 * rp[1] + w1 * rp[4] + w2 * rp[7];
    float pb = w0 * rp[2] + w1 * rp[5] + w2 * rp[8];
    float pd = w0 * m0 + w1 * m1 + w2 * m2;
    float pa = w0 + w1 + w2;
    if (lane != 31) {                                 // k0+3 == 127 only on lane 31
        pr += w3 * rp[9]; pg += w3 * rp[10]; pb += w3 * rp[11];
        pd += w3 * m3;    pa += w3;
    }

    #pragma unroll
    for (int off = 16; off > 0; off >>= 1) {
        pr += __shfl_xor(pr, off, 32);
        pg += __shfl_xor(pg, off, 32);
        pb += __shfl_xor(pb, off, 32);
        pd += __shfl_xor(pd, off, 32);
        pa += __shfl_xor(pa, off, 32);
    }

    if (lane == 0) {
        float* o = out + (size_t)ray * OUT_STRIDE;
        o[0] = pr; o[1] = pg; o[2] = pb;
        o[3] = pd / (pa + 1e-10f);
        o[4] = pa;
    }

    // normalized pdf (127 entries + zero pad) -> workspace (stays in 192MB L2)
    const float Sinv = 1.0f / (pa + 127.0f * 1e-5f);
    float4 pv;
    pv.x = (w0 + 1e-5f) * Sinv;
    pv.y = (w1 + 1e-5f) * Sinv;
    pv.z = (w2 + 1e-5f) * Sinv;
    pv.w = (lane == 31) ? 0.0f : (w3 + 1e-5f) * Sinv;
    *reinterpret_cast<float4*>(pdfw + (size_t)ray * N_SAMPLES + k0) = pv;
}

// ---------------------------------------------------------------------------
// Kernel 2: one wave32 per 16 rays.
//   CDF = pdf x StrictUpperTriangularOnes via V_WMMA_F32_16X16X4_F32,
//   z_vals staged to LDS with async global->LDS b128 copies,
//   then inverse-CDF sampling (binary search in LDS) -> d_fine.
// ---------------------------------------------------------------------------
__global__ __launch_bounds__(64) void k_sample(
    const float* __restrict__ pdfw, const float* __restrict__ zv,
    float* __restrict__ out)
{
    __shared__ __align__(16) float smem[2 * 4096];   // per wave: cdf[16][128] + z[16][128]
    const int wave = threadIdx.x >> 5;
    const int lane = threadIdx.x & 31;
    const int rowbase = (blockIdx.x * 2 + wave) * 16;

    float* lds_cdf = smem + wave * 4096;
    float* lds_z   = smem + wave * 4096 + 2048;

    // ---- async copy of 16 z_vals rows (8 KB) into LDS ----
    {
        const unsigned long long gbase =
            (unsigned long long)(uintptr_t)(zv + (size_t)rowbase * N_SAMPLES);
        const unsigned lbase = (unsigned)(uintptr_t)lds_z;   // low 32 bits = LDS offset
        #pragma unroll
        for (int i = 0; i < 16; ++i) {
            const int idx = i * 32 + lane;                   // float4 index
            unsigned long long ga = gbase + (unsigned long long)idx * 16u;
            unsigned la = lbase + (unsigned)idx * 16u;
            asm volatile("global_load_async_to_lds_b128 %0, %1, off"
                         :: "v"(la), "v"(ga) : "memory");
        }
    }

    // ---- WMMA triangular-matmul cumsum ----
    const int hf   = lane >> 4;                      // half-wave: K-pair select
    const int ncol = lane & 15;                      // output column within tile
    const int mrow = lane & 15;                      // A-fragment row (ray)

    v2f btri[4];
    #pragma unroll
    for (int j = 0; j < 4; ++j) {                    // strict k < j condition: 4j+kk < n
        int kk = 4 * j + 2 * hf;
        btri[j].x = (kk     < ncol) ? 1.0f : 0.0f;
        btri[j].y = (kk + 1 < ncol) ? 1.0f : 0.0f;
    }
    v2f bones; bones.x = 1.0f; bones.y = 1.0f;

    const float* prow = pdfw + (size_t)(rowbase + mrow) * N_SAMPLES;
    v8f P = {};                                      // running prefix, broadcast tile

    #pragma unroll
    for (int c = 0; c < 8; ++c) {
        v2f A[4];
        #pragma unroll
        for (int j = 0; j < 4; ++j) {                // A 16x4 f32 fragment layout
            const float2 p2 = *reinterpret_cast<const float2*>(
                prow + (4 * c + j) * 4 + 2 * hf);
            A[j].x = p2.x; A[j].y = p2.y;
        }
        v8f acc = P;
        #pragma unroll
        for (int j = 0; j < 4; ++j)
            acc = __builtin_amdgcn_wmma_f32_16x16x4_f32(
                false, A[j], false, btri[j], (short)0, acc, false, false);
        #pragma unroll
        for (int j = 0; j < 8; ++j) {                // D layout: M = j + 8*hf, N = ncol
            int m = j + hf * 8;
            lds_cdf[m * N_SAMPLES + c * 16 + ncol] = acc[j];
        }
        #pragma unroll
        for (int j = 0; j < 4; ++j)
            P = __builtin_amdgcn_wmma_f32_16x16x4_f32(
                false, A[j], false, bones, (short)0, P, false, false);
    }

    asm volatile("s_wait_asynccnt 0x0" ::: "memory");   // z rows resident in LDS

    // ---- inverse-CDF sampling: 16 rays x 64 samples = 32 items per lane ----
    for (int it = 0; it < 32; ++it) {
        const int item = it * 32 + lane;
        const int r  = item >> 6;
        const int si = item & 63;
        const float u = (float)si * (1.0f / 63.0f);
        const float* crow = lds_cdf + r * N_SAMPLES;

        int lo = 0, hi = N_SAMPLES;                  // searchsorted(..., side='right')
        while (lo < hi) {
            int mid = (lo + hi) >> 1;
            if (crow[mid] <= u) lo = mid + 1; else hi = mid;
        }
        int below = lo - 1;
        if (below < 0) below = 0;
        if (below > N_SAMPLES - 1) below = N_SAMPLES - 1;
        int above = (lo > N_SAMPLES - 1) ? (N_SAMPLES - 1) : lo;

        const float cg0 = crow[below], cg1 = crow[above];
        const float* zrow = lds_z + r * N_SAMPLES;
        const float b0 = zrow[below], b1 = zrow[above];
        float denom = cg1 - cg0;
        if (denom < 1e-5f) denom = 1.0f;
        const float t = (u - cg0) / denom;
        out[(size_t)(rowbase + r) * OUT_STRIDE + 5 + si] = b0 + t * (b1 - b0);
    }
}

// ---------------------------------------------------------------------------
extern "C" void kernel_launch(void* const* d_in, const int* in_sizes, int n_in,
                              void* d_out, int out_size, void* d_ws, size_t ws_size,
                              hipStream_t stream) {
    (void)in_sizes; (void)n_in; (void)out_size; (void)ws_size;
    const float* sdf = (const float*)d_in[0];
    const float* zv  = (const float*)d_in[1];
    const float* rad = (const float*)d_in[2];
    float* out  = (float*)d_out;
    float* pdfw = (float*)d_ws;                      // needs R*128*4 = 64 MB scratch

    k_weights<<<N_RAYS / 4, 128, 0, stream>>>(sdf, zv, rad, out, pdfw);
    k_sample <<<N_RAYS / 32, 64, 0, stream>>>(pdfw, zv, out);
}